// NGCF_90890097918586
// MI455X (gfx1250) — compile-verified
//
#include <hip/hip_runtime.h>

typedef __attribute__((ext_vector_type(2))) float v2f;
typedef __attribute__((ext_vector_type(8))) float v8f;

#define LEAKY_ALPHA 0.2f
#define L2_EPS 1e-12f
#define FD 64

// ---------------- utility kernels ----------------

__global__ void zero_f32_kernel(float* __restrict__ p, long long n) {
  long long i = (long long)blockIdx.x * blockDim.x + threadIdx.x;
  if (i < n) p[i] = 0.0f;
}

__global__ void degree_kernel(const int* __restrict__ rows, float* __restrict__ deg, int E) {
  int e = blockIdx.x * blockDim.x + threadIdx.x;
  if (e < E) atomicAdd(&deg[rows[e]], 1.0f);
}

__global__ void dinv_kernel(const float* __restrict__ deg, float* __restrict__ dinv, int N) {
  int n = blockIdx.x * blockDim.x + threadIdx.x;
  if (n < N) {
    float d = deg[n];
    dinv[n] = (d > 0.0f) ? rsqrtf(fmaxf(d, 1.0f)) : 0.0f;
  }
}

__global__ void copy_x_kernel(const float* __restrict__ x, float* __restrict__ out,
                              long long nd, int ostride) {
  long long t = (long long)blockIdx.x * blockDim.x + threadIdx.x;
  if (t < nd) {
    long long n = t >> 6;        // node
    int c = (int)(t & 63);       // column
    out[n * ostride + c] = x[t];
  }
}

// SpMM scatter: 16 threads per edge, 4 floats each.
// The float4 gathers of a row are contiguous across the 16 threads (coalesced),
// h (25.6MB) stays resident in the 192MB L2, atomics land in L2.
__global__ void spmm_kernel(const int* __restrict__ ei, const float* __restrict__ h,
                            const float* __restrict__ dinv, float* __restrict__ agg, int E) {
  long long t = (long long)blockIdx.x * blockDim.x + threadIdx.x;
  int e = (int)(t >> 4);
  if (e >= E) return;
  int sub = (int)(t & 15);
  int r = ei[e];
  int c = ei[(long long)E + e];
  float nm = dinv[r] * dinv[c];
  float4 m = *(const float4*)(h + (long long)c * FD + sub * 4);
  float* dst = agg + (long long)r * FD + sub * 4;
  atomicAdd(dst + 0, m.x * nm);
  atomicAdd(dst + 1, m.y * nm);
  atomicAdd(dst + 2, m.z * nm);
  atomicAdd(dst + 3, m.w * nm);
}

// Butterfly exchange across lanes with lane^mask via ds_swizzle_b32
// (group-of-32 mode: offset = xor_mask<<10 | and_mask 0x1f). EXEC is all-ones here.
__device__ __forceinline__ float bfly_xor1(float v) {
  return __int_as_float(__builtin_amdgcn_ds_swizzle(__float_as_int(v), (1 << 10) | 0x1f));
}
__device__ __forceinline__ float bfly_xor2(float v) {
  return __int_as_float(__builtin_amdgcn_ds_swizzle(__float_as_int(v), (2 << 10) | 0x1f));
}
__device__ __forceinline__ float bfly_xor4(float v) {
  return __int_as_float(__builtin_amdgcn_ds_swizzle(__float_as_int(v), (4 << 10) | 0x1f));
}
__device__ __forceinline__ float bfly_xor8(float v) {
  return __int_as_float(__builtin_amdgcn_ds_swizzle(__float_as_int(v), (8 << 10) | 0x1f));
}

// ---------------- fused dual-GEMM layer kernel (fp32 WMMA) ----------------
// h_new = leaky(agg@W1 + b1) + leaky((h*agg)@W2 + b2)
// hnext <- h_new (unnormalized, feeds next layer)
// out[:, col0:col0+64] <- l2_normalize(h_new)
//
// One wave: 16 nodes x 64 cols = 4 tiles of V_WMMA_F32_16X16X4_F32 x 2 GEMMs,
// K=64 fully unrolled (128 WMMAs). 8 waves/block -> 128 nodes/block.
//
// W is staged in LDS in "K-pair interleaved" form: pair (m, n) at float index
// m*128 + 2n holds {W[2m][n], W[2m+1][n]}, so a lane's B fragment
// {W[kb+2hi][n], W[kb+2hi+1][n]} is ONE even-aligned ds_load_b64 at an
// immediate offset (max 15744B < 64K) off a single per-lane base.
// A fragments (agg/h rows) are fully preloaded into static registers.
__global__ __launch_bounds__(256) void ngcf_layer_kernel(
    const float* __restrict__ agg, const float* __restrict__ hcur,
    const float* __restrict__ W1, const float* __restrict__ b1,
    const float* __restrict__ W2, const float* __restrict__ b2,
    float* __restrict__ hnext, float* __restrict__ out,
    int out_col0, int ostride, int N) {
  __shared__ __align__(16) float sW1[FD * FD];
  __shared__ __align__(16) float sW2[FD * FD];

  int tid = threadIdx.x;
  // Stage both matrices pair-interleaved: float4 {W[2m][n],W[2m+1][n],W[2m][n+1],W[2m+1][n+1]}
#pragma unroll
  for (int i = 0; i < 4; i++) {
    int idx = i * 256 + tid;     // 0..1023 float4 slots per matrix
    int m = idx >> 5;            // K-pair row (K = 2m, 2m+1), 0..31
    int n = (idx & 31) * 2;      // column 0,2,...,62
    float2 r0 = *(const float2*)&W1[(2 * m) * FD + n];
    float2 r1 = *(const float2*)&W1[(2 * m + 1) * FD + n];
    *(float4*)&sW1[m * 2 * FD + n * 2] = make_float4(r0.x, r1.x, r0.y, r1.y);
    float2 s0 = *(const float2*)&W2[(2 * m) * FD + n];
    float2 s1 = *(const float2*)&W2[(2 * m + 1) * FD + n];
    *(float4*)&sW2[m * 2 * FD + n * 2] = make_float4(s0.x, s1.x, s0.y, s1.y);
  }
  __syncthreads();

  int wave = tid >> 5;
  int lane = tid & 31;
  int lo = lane & 15;   // column / A-row index within tile
  int hi = lane >> 4;   // half-wave select (K pair for A/B, +8 row offset for C/D)
  int node0 = blockIdx.x * 128 + wave * 16;
  if (node0 >= N) return;  // whole-wave uniform (N % 16 == 0)

  v8f acc1[4], acc2[4];
#pragma unroll
  for (int t = 0; t < 4; t++)
#pragma unroll
    for (int r = 0; r < 8; r++) { acc1[t][r] = 0.0f; acc2[t][r] = 0.0f; }

  // Preload ALL A fragments (agg and h rows) up front: two global clauses with
  // immediate offsets; constant indices below keep these in static registers.
  const float* aRow = agg  + (long long)(node0 + lo) * FD + 2 * hi;
  const float* hRow = hcur + (long long)(node0 + lo) * FD + 2 * hi;
  v2f a1f[16], ahf[16];
#pragma unroll
  for (int j = 0; j < 16; j++) {
    a1f[j] = *(const v2f*)(aRow + 4 * j);
    ahf[j] = *(const v2f*)(hRow + 4 * j);
  }

  // Per-lane LDS base: pair (kb/2 + hi, t*16+lo) -> base + (kb/2)*128 + t*32 floats
  const float* bw1 = sW1 + hi * 2 * FD + lo * 2;
  const float* bw2 = sW2 + hi * 2 * FD + lo * 2;

  // FULL unroll: j is compile-time constant -> a1f/ahf stay in static VGPRs
  // (no v_movrels), all ds offsets are immediates.
#pragma unroll
  for (int j = 0; j < 16; j++) {
    const int kb = j * 4;
    v2f a1 = a1f[j];
    v2f a2 = a1 * ahf[j];  // elementwise gate for second GEMM (v_pk_mul_f32)
#pragma unroll
    for (int t = 0; t < 4; t++) {
      // B fragment: component v holds W[kb+2hi+v][t*16+lo]
      // (K-ordering assumed to mirror the documented f32 A 16x4 layout)
      v2f bf1 = *(const v2f*)(bw1 + (kb >> 1) * 2 * FD + t * 32);
      v2f bf2 = *(const v2f*)(bw2 + (kb >> 1) * 2 * FD + t * 32);
      acc1[t] = __builtin_amdgcn_wmma_f32_16x16x4_f32(
          false, a1, false, bf1, (short)0, acc1[t], false, false);
      acc2[t] = __builtin_amdgcn_wmma_f32_16x16x4_f32(
          false, a2, false, bf2, (short)0, acc2[t], false, false);
    }
  }

  // Epilogue: bias + leaky + sum
  float hval[4][8];
#pragma unroll
  for (int t = 0; t < 4; t++) {
    float bias1 = b1[t * 16 + lo];
    float bias2 = b2[t * 16 + lo];
#pragma unroll
    for (int r = 0; r < 8; r++) {
      float z1 = acc1[t][r] + bias1;
      z1 = (z1 > 0.0f) ? z1 : LEAKY_ALPHA * z1;
      float z2 = acc2[t][r] + bias2;
      z2 = (z2 > 0.0f) ? z2 : LEAKY_ALPHA * z2;
      hval[t][r] = z1 + z2;
    }
  }

  // Row-wise l2 norm: ds_swizzle butterfly over the 16 lanes sharing `hi`
  // (masks < 16 never cross the half-wave, so `hi` groups stay independent).
#pragma unroll
  for (int r = 0; r < 8; r++) {
    float sq = 0.0f;
#pragma unroll
    for (int t = 0; t < 4; t++) sq += hval[t][r] * hval[t][r];
    sq += bfly_xor1(sq);
    sq += bfly_xor2(sq);
    sq += bfly_xor4(sq);
    sq += bfly_xor8(sq);
    float rn = rsqrtf(fmaxf(sq, L2_EPS));
    long long node = node0 + r + 8 * hi;  // C/D layout: component r -> row r + 8*hi
    float* hn = hnext + node * FD;
    float* op = out + node * ostride + out_col0;
#pragma unroll
    for (int t = 0; t < 4; t++) {
      float v = hval[t][r];
      hn[t * 16 + lo] = v;       // unnormalized -> next layer
      op[t * 16 + lo] = v * rn;  // normalized -> output slice
    }
  }
}

// ---------------- launcher ----------------

extern "C" void kernel_launch(void* const* d_in, const int* in_sizes, int n_in,
                              void* d_out, int out_size, void* d_ws, size_t ws_size,
                              hipStream_t stream) {
  const float* x  = (const float*)d_in[0];
  const int*   ei = (const int*)d_in[1];
  const float* W1 = (const float*)d_in[2];
  const float* b1 = (const float*)d_in[3];
  const float* W2 = (const float*)d_in[4];
  const float* b2 = (const float*)d_in[5];
  float* out = (float*)d_out;

  const int N = in_sizes[0] / FD;
  const int E = in_sizes[1] / 2;
  const int K = in_sizes[2] / (FD * FD);
  const int ostride = FD * (K + 1);

  // workspace carve-up (256B-aligned regions)
  float* ws = (float*)d_ws;
  size_t off = 0;
  auto alignup = [](size_t v) { return (v + 63) & ~(size_t)63; };
  float* deg  = ws + off; off = alignup(off + (size_t)N);
  float* dinv = ws + off; off = alignup(off + (size_t)N);
  float* aggb = ws + off; off = alignup(off + (size_t)N * FD);
  float* hA   = ws + off; off = alignup(off + (size_t)N * FD);
  float* hB   = ws + off;

  dim3 blk(256);
  long long nd = (long long)N * FD;

  // degree -> dinv
  zero_f32_kernel<<<(N + 255) / 256, blk, 0, stream>>>(deg, (long long)N);
  degree_kernel<<<(E + 255) / 256, blk, 0, stream>>>(ei, deg, E);
  dinv_kernel<<<(N + 255) / 256, blk, 0, stream>>>(deg, dinv, N);

  // out[:, :64] = x
  copy_x_kernel<<<(unsigned)((nd + 255) / 256), blk, 0, stream>>>(x, out, nd, ostride);

  const float* hcur = x;
  float* bufs[2] = {hA, hB};
  for (int i = 0; i < K; i++) {
    zero_f32_kernel<<<(unsigned)((nd + 255) / 256), blk, 0, stream>>>(aggb, nd);
    long long spthreads = (long long)E * 16;
    spmm_kernel<<<(unsigned)((spthreads + 255) / 256), blk, 0, stream>>>(ei, hcur, dinv, aggb, E);
    ngcf_layer_kernel<<<(N + 127) / 128, blk, 0, stream>>>(
        aggb, hcur,
        W1 + (size_t)i * FD * FD, b1 + (size_t)i * FD,
        W2 + (size_t)i * FD * FD, b2 + (size_t)i * FD,
        bufs[i & 1], out, FD + i * FD, ostride, N);
    hcur = bufs[i & 1];
  }
}